// BioClassifier_90314572300884
// MI455X (gfx1250) — compile-verified
//
#include <hip/hip_runtime.h>
#include <hip/hip_bf16.h>

// ---------------------------------------------------------------------------
// BioClassifier (whiten -> sequential Oja scan -> ReLU readout) for gfx1250
//   B=2048, D=784, H=256, O=10, LR=1e-3
// ---------------------------------------------------------------------------

typedef __attribute__((ext_vector_type(2))) float    v2f;
typedef __attribute__((ext_vector_type(8))) float    v8f;
typedef __attribute__((ext_vector_type(4))) unsigned v4u;
typedef __attribute__((ext_vector_type(8))) int      v8i_;
typedef __attribute__((ext_vector_type(4))) int      v4i_;

#define BB   2048
#define DD   784
#define HH   256
#define OO   10
#define LRc  1e-3f

#define NWG  4          // cooperative workgroups in the Oja scan
#define DSL  (DD / NWG) // 196 columns of W per workgroup
#define WST  197        // LDS row stride (gcd(197,64)=1 -> bank-conflict-free)
#define XLP  200        // x-slice buffer pitch (floats)

// ---------------------------------------------------------------------------
// Tensor Data Mover helpers: 1-row tile (nelem contiguous f32) between global
// and LDS.  D# per ISA §8: group0 = {flags, lds_addr, global_addr, type=2},
// group1 = {data_size=4B, tensor_dim0, tensor_dim1=1, tile_dim0, tile_dim1=1,
// tensor_dim0_stride}.  Tracked by TENSORcnt.
// ---------------------------------------------------------------------------
__device__ __forceinline__ void tdm_desc(v4u& g0, v8i_& g1,
                                         const void* lds_ptr, const void* gptr,
                                         unsigned nelem, unsigned row_stride) {
  const unsigned long long ga = (unsigned long long)gptr;
  g0.x = 1u;                                   // count=1, user descriptor
  g0.y = (unsigned)(unsigned long long)lds_ptr; // LDS byte offset (addr[31:0])
  g0.z = (unsigned)ga;                          // global_addr[31:0]
  g0.w = ((unsigned)(ga >> 32) & 0x01FFFFFFu)   // global_addr[56:32]
         | (2u << 30);                          // D#.type = 2
  g1[0] = 0x20000;                              // data_size code 2 (4 bytes)
  g1[1] = (int)(nelem << 16);                   // tensor_dim0[15:0]
  g1[2] = (int)(1u << 16);                      // tensor_dim1 = 1
  g1[3] = (int)(nelem << 16);                   // tile_dim0 = nelem
  g1[4] = 1;                                    // tile_dim1 = 1
  g1[5] = (int)row_stride;                      // tensor_dim0_stride (elems)
  g1[6] = 0;
  g1[7] = 0;
}

__device__ __forceinline__ void tdm_load_row(const void* lds_ptr, const void* gptr,
                                             unsigned nelem, unsigned stride) {
  v4u g0; v8i_ g1;
  v4i_ gz = {0, 0, 0, 0};
  v8i_ gz8 = {0, 0, 0, 0, 0, 0, 0, 0};
  tdm_desc(g0, g1, lds_ptr, gptr, nelem, stride);
  __builtin_amdgcn_tensor_load_to_lds(g0, g1, gz, gz, gz8, 0);
}

__device__ __forceinline__ void tdm_store_row(const void* lds_ptr, void* gptr,
                                              unsigned nelem, unsigned stride) {
  v4u g0; v8i_ g1;
  v4i_ gz = {0, 0, 0, 0};
  v8i_ gz8 = {0, 0, 0, 0, 0, 0, 0, 0};
  tdm_desc(g0, g1, lds_ptr, gptr, nelem, stride);
  __builtin_amdgcn_tensor_store_from_lds(g0, g1, gz, gz, gz8, 0);
}

// ---------------------------------------------------------------------------
// Kernel 1: whitening GEMM  xw[B,D] = (x - mu) @ M^T   (exact f32 via WMMA)
// grid (16, 49), block 256 (8 waves); wave w -> one 16x16 tile.
// ---------------------------------------------------------------------------
__global__ void __launch_bounds__(256)
whiten_wmma(const float* __restrict__ x, const float* __restrict__ mu,
            const float* __restrict__ Wm, float* __restrict__ xw) {
  const int wave = threadIdx.x >> 5;
  const int lane = threadIdx.x & 31;
  const int half = lane >> 4;       // 0: K=+0,+1 ; 1: K=+2,+3 (16x4 f32 A layout)
  const int l16  = lane & 15;
  const int mtile = blockIdx.x * 8 + wave;
  const int ntile = blockIdx.y;
  const int m = mtile * 16 + l16;   // A row held by this lane
  const int n = ntile * 16 + l16;   // B column held by this lane

  const float* __restrict__ xrow = x  + (size_t)m * DD;
  const float* __restrict__ wrow = Wm + (size_t)n * DD;  // B[k][n] = M[n][k]

  v8f c = {};
  #pragma unroll 4
  for (int kb = 0; kb < DD; kb += 4) {
    const int kk = kb + 2 * half;
    v2f a, b;
    a.x = xrow[kk]     - mu[kk];
    a.y = xrow[kk + 1] - mu[kk + 1];
    b.x = wrow[kk];
    b.y = wrow[kk + 1];
    c = __builtin_amdgcn_wmma_f32_16x16x4_f32(false, a, false, b,
                                              (short)0, c, false, false);
  }
  const int rbase = mtile * 16 + half * 8;
  #pragma unroll
  for (int r = 0; r < 8; ++r)
    xw[(size_t)(rbase + r) * DD + ntile * 16 + l16] = c[r];
}

// ---------------------------------------------------------------------------
// Kernel 2: reset the cooperative-barrier counter (graph-replay determinism).
// ---------------------------------------------------------------------------
__global__ void scan_init(unsigned* barcnt) { *barcnt = 0u; }

// ---------------------------------------------------------------------------
// Grid-wide barrier over NWG workgroups (monotonic counter, absolute target).
// ---------------------------------------------------------------------------
__device__ __forceinline__ void gbar(unsigned* cnt, unsigned target) {
  __threadfence();                      // release our global writes
  __syncthreads();
  if (threadIdx.x == 0) {
    atomicAdd(cnt, 1u);
    while (*(volatile unsigned*)cnt < target)
      __builtin_amdgcn_s_sleep(2);
  }
  __syncthreads();
  __threadfence();                      // acquire others' writes
}

// ---------------------------------------------------------------------------
// Kernel 3: sequential Oja scan.  4 cooperative WGs, each owns a 256x196
// column slice of W, LDS-resident (stride 197 -> conflict-free both sweeps).
// Per sample i:
//   y_i = W_i x_i            (partial y was produced by the previous fused
//                             sweep; cross-WG reduce via L2 + grid barrier)
//   u   = x_i - W_i^T y_i    (local-column sweep)
//   W  += lr * y_i u^T  FUSED with  partial y_{i+1} = W_{i+1} x_{i+1}
// x slices are double-buffered via TDM (tensor_load_to_lds) one sample ahead;
// feature rows leave LDS via TDM (tensor_store_from_lds) from WG0.
// ---------------------------------------------------------------------------
__global__ void __launch_bounds__(256)
oja_scan(const float* __restrict__ xw, const float* __restrict__ W0,
         float* __restrict__ feats, float* __restrict__ ypart,
         unsigned* __restrict__ barcnt) {
  extern __shared__ float smem[];
  float* Wl = smem;                 // [256][197]
  float* xl = Wl + HH * WST;        // [2][XLP] double-buffered x slice
  float* uu = xl + 2 * XLP;         // [XLP]  x - W^T y
  float* yy = uu + XLP;             // [256]  reduced y

  const int t  = threadIdx.x;       // 256 threads (8 waves)
  const int g  = blockIdx.x;        // slice id
  const int c0 = g * DSL;
  // scalarized wave id -> s_cbranch around TDM ops (TDM ignores EXEC!)
  const int wid = __builtin_amdgcn_readfirstlane((int)(threadIdx.x >> 5));

  // Kick off the TDM fetch of sample 0's slice while we stage W.
  if (wid == 0)
    tdm_load_row(xl, xw + c0, DSL, DD);

  for (int idx = t; idx < HH * DSL; idx += 256) {
    const int h = idx / DSL;
    const int d = idx - h * DSL;
    Wl[h * WST + d] = W0[(size_t)h * DD + c0 + d];
  }
  if (wid == 0) __builtin_amdgcn_s_wait_tensorcnt(0);
  __syncthreads();

  // Prologue: partial y for sample 0 (thread = row, conflict-free row sweep).
  {
    const float* wr = &Wl[t * WST];
    float s = 0.f;
    #pragma unroll 4
    for (int d = 0; d < DSL; ++d) s = __builtin_fmaf(wr[d], xl[d], s);
    ypart[g * HH + t] = s;
  }

  unsigned bar = 0;
  for (int i = 0; i < BB; ++i) {
    const float* xcur = xl + (i & 1) * XLP;
    float*       xnxt = xl + ((i + 1) & 1) * XLP;

    // Prefetch sample i+1 via TDM (buffer's last reader was fenced at i-1).
    if (wid == 0 && i + 1 < BB)
      tdm_load_row(xnxt, xw + (size_t)(i + 1) * DD + c0, DSL, DD);

    gbar(barcnt, ++bar * NWG);          // partial y visible

    // Reduce y across the NWG slices.
    {
      float s = 0.f;
      #pragma unroll
      for (int w = 0; w < NWG; ++w) s += ypart[w * HH + t];
      yy[t] = s;
    }
    gbar(barcnt, ++bar * NWG);          // all reads done before next overwrite

    // WG0 streams the (pre-update) feature row out of LDS via TDM.
    if (g == 0 && wid == 0)
      tdm_store_row(yy, feats + (size_t)i * HH, HH, HH);

    // u_d = x_d - sum_h y_h W[h,d]  (thread = local column, stride-197 sweep)
    if (t < DSL) {
      float s = 0.f;
      #pragma unroll 4
      for (int h = 0; h < HH; ++h) s = __builtin_fmaf(yy[h], Wl[h * WST + t], s);
      uu[t] = xcur[t] - s;
    }

    // x_{i+1} landed; also retires WG0's feats store before yy is rewritten.
    if (wid == 0) __builtin_amdgcn_s_wait_tensorcnt(0);
    __syncthreads();

    // Fused sweep: rank-1 update of W and partial y for sample i+1.
    if (i + 1 < BB) {
      const float yl = LRc * yy[t];
      float* wr = &Wl[t * WST];
      float s2 = 0.f;
      #pragma unroll 4
      for (int d = 0; d < DSL; ++d) {
        const float w2 = __builtin_fmaf(yl, uu[d], wr[d]);
        wr[d] = w2;
        s2 = __builtin_fmaf(w2, xnxt[d], s2);
      }
      ypart[g * HH + t] = s2;
    }
  }
}

// ---------------------------------------------------------------------------
// Kernel 4: logits = relu(feats) @ R^T + b  via f32 WMMA, N padded 10->16.
// ---------------------------------------------------------------------------
__global__ void __launch_bounds__(256)
readout_wmma(const float* __restrict__ feats, const float* __restrict__ rW,
             const float* __restrict__ rb, float* __restrict__ out) {
  const int wave = threadIdx.x >> 5;
  const int lane = threadIdx.x & 31;
  const int half = lane >> 4;
  const int l16  = lane & 15;
  const int mtile = blockIdx.x * 8 + wave;
  const int m = mtile * 16 + l16;

  const float nmask = (l16 < OO) ? 1.0f : 0.0f;   // branchless N padding
  const int   nc    = (l16 < OO) ? l16 : 0;

  const float* __restrict__ frow = feats + (size_t)m * HH;
  const float* __restrict__ wrow = rW + (size_t)nc * HH;

  v8f c = {};
  #pragma unroll 4
  for (int kb = 0; kb < HH; kb += 4) {
    const int kk = kb + 2 * half;
    v2f a, b;
    a.x = fmaxf(frow[kk], 0.f);          // fused ReLU on the A fragment
    a.y = fmaxf(frow[kk + 1], 0.f);
    b.x = wrow[kk] * nmask;
    b.y = wrow[kk + 1] * nmask;
    c = __builtin_amdgcn_wmma_f32_16x16x4_f32(false, a, false, b,
                                              (short)0, c, false, false);
  }
  if (l16 < OO) {                        // divergence only after the WMMAs
    const float bias = rb[l16];
    const int rbase = mtile * 16 + half * 8;
    #pragma unroll
    for (int r = 0; r < 8; ++r)
      out[(size_t)(rbase + r) * OO + l16] = c[r] + bias;
  }
}

// ---------------------------------------------------------------------------
extern "C" void kernel_launch(void* const* d_in, const int* in_sizes, int n_in,
                              void* d_out, int out_size, void* d_ws, size_t ws_size,
                              hipStream_t stream) {
  const float* x   = (const float*)d_in[0];  // [2048,784]
  const float* mu  = (const float*)d_in[1];  // [784]
  const float* Wm  = (const float*)d_in[2];  // [784,784]
  const float* W0  = (const float*)d_in[3];  // [256,784]
  const float* rW  = (const float*)d_in[4];  // [10,256]
  const float* rb  = (const float*)d_in[5];  // [10]
  float* logits    = (float*)d_out;          // [2048,10]

  // workspace layout
  float*    xw     = (float*)d_ws;                   // 2048*784
  float*    feats  = xw + (size_t)BB * DD;           // 2048*256
  float*    ypart  = feats + (size_t)BB * HH;        // NWG*256
  unsigned* barcnt = (unsigned*)(ypart + NWG * HH);  // 1

  whiten_wmma<<<dim3(BB / 128, DD / 16), 256, 0, stream>>>(x, mu, Wm, xw);
  scan_init<<<1, 1, 0, stream>>>(barcnt);
  const size_t smem = (size_t)(HH * WST + 2 * XLP + XLP + HH) * sizeof(float);
  oja_scan<<<NWG, 256, smem, stream>>>(xw, W0, feats, ypart, barcnt);
  readout_wmma<<<BB / 128, 256, 0, stream>>>(feats, rW, rb, logits);
}